// GCN_14310831030615
// MI455X (gfx1250) — compile-verified
//
#include <hip/hip_runtime.h>

// ---------------------------------------------------------------------------
// GCN forward on MI455X (gfx1250, wave32).
//  - Dense transforms: V_WMMA_F32_16X16X4_F32, A-tile staged per block into
//    LDS via GLOBAL_LOAD_ASYNC_TO_LDS_B128 (ASYNCcnt) -> ds_load fragments.
//  - Message passing: one wave32 per edge, float4 gather, f32 global atomics,
//    global_prefetch of the next edge's gather row.
//  - d_out initialized with bias so scatter atomics land directly on it.
// ---------------------------------------------------------------------------

#define IN_C 128
#define HID  128
#define OUT_C 64
#define EPS  1e-5f

typedef __attribute__((ext_vector_type(2))) float v2f;
typedef __attribute__((ext_vector_type(8))) float v8f;

__device__ __forceinline__ void atomAddF(float* p, float v) {
  __hip_atomic_fetch_add(p, v, __ATOMIC_RELAXED, __HIP_MEMORY_SCOPE_AGENT);
}

// ----------------------------- utility fills -------------------------------
__global__ void zero4_kernel(float4* __restrict__ p, size_t n4) {
  size_t i = blockIdx.x * (size_t)blockDim.x + threadIdx.x;
  size_t stride = (size_t)gridDim.x * blockDim.x;
  float4 z = make_float4(0.f, 0.f, 0.f, 0.f);
  for (; i < n4; i += stride) p[i] = z;
}

// Initialize an accumulator surface with a broadcast per-channel bias.
__global__ void fill_bias(float* __restrict__ out, const float* __restrict__ bias,
                          int mask, size_t total) {
  size_t i = blockIdx.x * (size_t)blockDim.x + threadIdx.x;
  size_t stride = (size_t)gridDim.x * blockDim.x;
  for (; i < total; i += stride) out[i] = bias[(int)(i & (size_t)mask)];
}

// ------------------------------ degree / norm ------------------------------
__global__ void degree_kernel(const int* __restrict__ dst, float* __restrict__ deg, int E) {
  int i = blockIdx.x * blockDim.x + threadIdx.x;
  if (i < E) atomAddF(&deg[dst[i]], 1.0f);
}

__global__ void dinv_kernel(float* __restrict__ deg, int n) {
  int i = blockIdx.x * blockDim.x + threadIdx.x;
  if (i < n) {
    float d = deg[i];
    deg[i] = (d > 0.0f) ? rsqrtf(fmaxf(d, 1.0f)) : 0.0f;  // in place: deg -> deg^{-1/2}
  }
}

// ------------------------------- WMMA GEMM ---------------------------------
// C[M x NCOLS] = A[M x 128] * W[128 x NCOLS], fp32 row-major.
// Block = one 16-row stripe; NCOLS/16 waves, one 16x16 tile per wave.
// The 16x128 A tile is a contiguous 8KB region: async-DMA it into LDS once
// per block (GLOBAL_LOAD_ASYNC_TO_LDS_B128), then all waves read fragments
// from LDS. K=128 -> 32 x V_WMMA_F32_16X16X4_F32 per wave.
//
// Fragment layouts (ISA 7.12.2, wave32):
//   A 16x4 (2 VGPRs): lanes 0-15 row M=lane  -> {K=k, K=k+1}
//                     lanes 16-31 row M=lane-16 -> {K=k+2, K=k+3}
//   B 4x16 (2 VGPRs): VGPR0 = rows {k, k+2}, VGPR1 = rows {k+1, k+3}
//   C/D 16x16 (8 VGPRs): VGPR v: lanes 0-15 -> M=v, lanes 16-31 -> M=v+8
template <int NCOLS>
__global__ void gemm_wmma_f32(const float* __restrict__ A,
                              const float* __restrict__ W,
                              float* __restrict__ C) {
  __shared__ float tileA[16 * 128];            // 8 KB
  const int tid   = threadIdx.x;
  const int wave  = tid >> 5;
  const int lane  = tid & 31;
  const int lhalf = lane & 15;
  const bool hi   = lane >= 16;
  const int m0 = blockIdx.x * 16;
  const int n0 = wave * 16;
  constexpr int NTHREADS = NCOLS * 2;          // 256 (NCOLS=128) or 128 (NCOLS=64)
  constexpr int CHUNKS   = (16 * 128 * 4) / 16; // 512 x 16B transfers

  // --- async stage of the contiguous 8KB A tile into LDS ---
  {
    const unsigned lbase = (unsigned)(unsigned long long)tileA;  // LDS byte offset
    const unsigned long long gbase =
        (unsigned long long)(const void*)(A + (size_t)m0 * 128);
#pragma unroll
    for (int c = tid; c < CHUNKS; c += NTHREADS) {
      const unsigned l = lbase + (unsigned)c * 16u;
      const unsigned long long g = gbase + (unsigned long long)c * 16ull;
      asm volatile("global_load_async_to_lds_b128 %0, %1, off"
                   :: "v"(l), "v"(g) : "memory");
    }
    asm volatile("s_wait_asynccnt 0x0" ::: "memory");
    __syncthreads();
  }

  const float* arow = tileA + lhalf * 128;     // ds_load source
  const int kofs = hi ? 2 : 0;

  v8f acc = {};
#pragma unroll 4
  for (int k = 0; k < 128; k += 4) {
    v2f a, b;
    const int ka = k + kofs;
    a.x = arow[ka];
    a.y = arow[ka + 1];
    b.x = W[(size_t)ka * NCOLS + n0 + lhalf];
    b.y = W[(size_t)(ka + 1) * NCOLS + n0 + lhalf];
    acc = __builtin_amdgcn_wmma_f32_16x16x4_f32(
        /*neg_a=*/false, a, /*neg_b=*/false, b,
        /*c_mod=*/(short)0, acc, /*reuse_a=*/false, /*reuse_b=*/false);
  }

  const int col   = n0 + lhalf;
  const int rbase = m0 + (hi ? 8 : 0);
#pragma unroll
  for (int v = 0; v < 8; ++v)
    C[(size_t)(rbase + v) * NCOLS + col] = acc[v];
}

// --------------------------- edge scatter (GCN) ----------------------------
// agg[dst] += H[src] * dinv[src]*dinv[dst].  One wave per edge, grid-stride;
// vectorized row gather (b128/b64 per lane), per-element f32 atomics.
template <int WIDTH>
__global__ void edge_scatter(const float* __restrict__ H,
                             const float* __restrict__ dinv,
                             const int* __restrict__ src,
                             const int* __restrict__ dst,
                             float* __restrict__ agg, int E) {
  const int lane = threadIdx.x & 31;
  int wid = (int)((blockIdx.x * (size_t)blockDim.x + threadIdx.x) >> 5);
  const int nwaves = (int)(((size_t)gridDim.x * blockDim.x) >> 5);
  constexpr int PER = WIDTH / 32;              // 4 or 2 floats per lane

  for (int e = wid; e < E; e += nwaves) {
    const int s = src[e];
    const int d = dst[e];
    const int en = e + nwaves;                 // uniform per wave
    if (en < E) {
      const int sn = src[en];
      __builtin_prefetch(H + (size_t)sn * WIDTH + lane * PER, 0, 0);
    }
    const float nrm = dinv[s] * dinv[d];
    const float* hs = H + (size_t)s * WIDTH;
    float* ad = agg + (size_t)d * WIDTH + lane * PER;
    if constexpr (PER == 4) {
      const float4 v = ((const float4*)hs)[lane];   // global_load_b128
      atomAddF(ad + 0, v.x * nrm);
      atomAddF(ad + 1, v.y * nrm);
      atomAddF(ad + 2, v.z * nrm);
      atomAddF(ad + 3, v.w * nrm);
    } else {
      const float2 v = ((const float2*)hs)[lane];   // global_load_b64
      atomAddF(ad + 0, v.x * nrm);
      atomAddF(ad + 1, v.y * nrm);
    }
  }
}

// ------------------------------- BatchNorm ---------------------------------
// stats[0..127] = sum, stats[128..255] = sumsq (atomically accumulated).
__global__ void bn_stats(const float* __restrict__ h, float* __restrict__ stats, int M) {
  const int c = threadIdx.x;  // 128 threads = one channel each
  float s1 = 0.f, s2 = 0.f;
  for (int r = blockIdx.x; r < M; r += gridDim.x) {
    const float v = h[(size_t)r * 128 + c];
    s1 += v;
    s2 += v * v;
  }
  atomAddF(&stats[c], s1);
  atomAddF(&stats[128 + c], s2);
}

// scale/shift into stats[256..383] / stats[384..511].
// (Conv bias b1 cancels exactly under mean subtraction -> omitted.)
__global__ void bn_finalize(float* __restrict__ stats,
                            const float* __restrict__ gamma,
                            const float* __restrict__ beta, float invM) {
  const int c = threadIdx.x;
  const float mean = stats[c] * invM;
  const float var  = stats[128 + c] * invM - mean * mean;
  const float is   = rsqrtf(var + EPS);
  const float sc   = gamma[c] * is;
  stats[256 + c] = sc;
  stats[384 + c] = beta[c] - mean * sc;
}

// In-place BN + ReLU over [M x 128].
__global__ void bn_apply_relu(float* __restrict__ h, const float* __restrict__ scsh,
                              size_t total) {
  size_t i = blockIdx.x * (size_t)blockDim.x + threadIdx.x;
  size_t stride = (size_t)gridDim.x * blockDim.x;
  for (; i < total; i += stride) {
    const int c = (int)(i & 127);
    h[i] = fmaxf(h[i] * scsh[c] + scsh[128 + c], 0.0f);
  }
}

// ---------------------------------------------------------------------------
extern "C" void kernel_launch(void* const* d_in, const int* in_sizes, int n_in,
                              void* d_out, int out_size, void* d_ws, size_t ws_size,
                              hipStream_t stream) {
  const float* x     = (const float*)d_in[0];
  const int*   ei    = (const int*)  d_in[1];
  const float* W1    = (const float*)d_in[2];
  // d_in[3] = b1: cancels exactly under BatchNorm mean subtraction.
  const float* gamma = (const float*)d_in[4];
  const float* beta  = (const float*)d_in[5];
  const float* Wf    = (const float*)d_in[6];
  const float* bf    = (const float*)d_in[7];
  const float* Wo    = (const float*)d_in[8];
  const float* bo    = (const float*)d_in[9];

  const int n = in_sizes[0] / IN_C;   // 50000 (divisible by 16)
  const int E = in_sizes[1] / 2;      // 600000
  const int* srcv = ei;
  const int* dstv = ei + E;

  // Workspace layout (floats): dinv | stats(512) | h1/hf | agg1/hbn | ho
  float* ws    = (float*)d_ws;
  float* dinv  = ws;                               // n
  float* stats = dinv + n;                         // 512
  float* h1    = stats + 512;                      // n*128 (reused as hf)
  float* agg1  = h1 + (size_t)n * HID;             // n*128 (BN applied in place)
  float* ho    = agg1 + (size_t)n * HID;           // n*64
  const size_t need = ((size_t)n + 512 + (size_t)n * (HID + HID + OUT_C)) * sizeof(float);
  if (ws_size < need) return;

  float* outp  = (float*)d_out;
  float* feat  = outp;                             // n*128
  float* logit = outp + (size_t)n * IN_C;          // n*64

  // --- init accumulators: ws zeros; d_out pre-loaded with biases so the   ---
  // --- head scatters accumulate straight into the final output buffers.   ---
  zero4_kernel<<<256, 256, 0, stream>>>((float4*)dinv, (size_t)n / 4);
  zero4_kernel<<<1, 128, 0, stream>>>((float4*)stats, 512 / 4);
  zero4_kernel<<<4096, 256, 0, stream>>>((float4*)agg1, (size_t)n * HID / 4);
  fill_bias<<<2048, 256, 0, stream>>>(feat, bf, IN_C - 1, (size_t)n * IN_C);
  fill_bias<<<1024, 256, 0, stream>>>(logit, bo, OUT_C - 1, (size_t)n * OUT_C);

  // --- gcn_norm: degree -> deg^{-1/2} (in place) ---
  degree_kernel<<<(E + 255) / 256, 256, 0, stream>>>(dstv, dinv, E);
  dinv_kernel<<<(n + 255) / 256, 256, 0, stream>>>(dinv, n);

  // --- layer 1: h1 = x @ W1 ; scatter into agg1 ---
  gemm_wmma_f32<HID><<<n / 16, 256, 0, stream>>>(x, W1, h1);
  edge_scatter<HID><<<2048, 256, 0, stream>>>(h1, dinv, srcv, dstv, agg1, E);

  // --- BatchNorm (training stats) + ReLU, in place on agg1 ---
  bn_stats<<<512, 128, 0, stream>>>(agg1, stats, n);
  bn_finalize<<<1, 128, 0, stream>>>(stats, gamma, beta, 1.0f / (float)n);
  bn_apply_relu<<<4096, 256, 0, stream>>>(agg1, stats + 256, (size_t)n * HID);

  // --- heads: hf = hbn @ Wf ; ho = hbn @ Wo ---
  gemm_wmma_f32<IN_C><<<n / 16, 256, 0, stream>>>(agg1, Wf, h1);   // reuse h1 as hf
  gemm_wmma_f32<OUT_C><<<n / 16, 128, 0, stream>>>(agg1, Wo, ho);

  // --- scatter heads directly onto bias-initialized d_out ---
  edge_scatter<IN_C><<<2048, 256, 0, stream>>>(h1, dinv, srcv, dstv, feat, E);
  edge_scatter<OUT_C><<<2048, 256, 0, stream>>>(ho, dinv, srcv, dstv, logit, E);
}